// SimplePredictor_69973607186859
// MI455X (gfx1250) — compile-verified
//
#include <hip/hip_runtime.h>
#include <math.h>

typedef __attribute__((ext_vector_type(2))) float v2f;
typedef __attribute__((ext_vector_type(4))) float v4f;
typedef __attribute__((ext_vector_type(8))) float v8f;

#define B_DIM   16
#define S_DIM   2048
#define D_DIM   1024
#define OUT_DIM 1024
#define L_DIM   4096
#define T_DIM   512
#define ZSPLIT  8           // S-dimension split for the pooling reduction
#define SCHUNK  (S_DIM / ZSPLIT)

// ---------------------------------------------------------------------------
// Kernel 1a: partial masked sums over an S-chunk.  512 blocks in flight so the
// 128 MB stream can actually saturate 23.3 TB/s.
// ---------------------------------------------------------------------------
__global__ __launch_bounds__(256) void pool_partial(
    const float* __restrict__ emb, const unsigned char* __restrict__ mask,
    float* __restrict__ part, float* __restrict__ pcnt) {
  const int d = blockIdx.x * 256 + threadIdx.x;
  const int b = blockIdx.y;
  const int z = blockIdx.z;
  const int s0 = z * SCHUNK;
  const float* eb = emb + ((size_t)b * S_DIM + s0) * D_DIM + d;
  const unsigned char* mb = mask + (size_t)b * S_DIM + s0;
  float acc = 0.f, cnt = 0.f;
  for (int s = 0; s < SCHUNK; ++s) {
    const float mv = mb[s] ? 1.f : 0.f;
    acc = fmaf(mv, eb[(size_t)s * D_DIM], acc);
    cnt += mv;
  }
  part[((size_t)z * B_DIM + b) * D_DIM + d] = acc;
  if (blockIdx.x == 0 && threadIdx.x == 0) pcnt[z * B_DIM + b] = cnt;
}

// ---------------------------------------------------------------------------
// Kernel 1b: deterministic fixed-order reduction of the 8 partials.
// ---------------------------------------------------------------------------
__global__ __launch_bounds__(256) void pool_reduce(
    const float* __restrict__ part, const float* __restrict__ pcnt,
    float* __restrict__ pooled) {
  const int d = blockIdx.x * 256 + threadIdx.x;
  const int b = blockIdx.y;
  float acc = 0.f, cnt = 0.f;
#pragma unroll
  for (int z = 0; z < ZSPLIT; ++z) {
    acc += part[((size_t)z * B_DIM + b) * D_DIM + d];
    cnt += pcnt[z * B_DIM + b];
  }
  pooled[b * D_DIM + d] = acc / fmaxf(cnt, 1.f);
}

// ---------------------------------------------------------------------------
// Kernel 2: base = pooled @ W1_top   (16 x 1024 x 1024 = 33 MFLOP, trivial)
// ---------------------------------------------------------------------------
__global__ __launch_bounds__(256) void base_kernel(
    const float* __restrict__ pooled, const float* __restrict__ W1top,
    float* __restrict__ base) {
  const int o = blockIdx.x * 256 + threadIdx.x;
  const int b = blockIdx.y;
  const float* pr = pooled + b * D_DIM;
  float acc = 0.f;
  for (int k = 0; k < D_DIM; ++k)
    acc = fmaf(pr[k], W1top[(size_t)k * OUT_DIM + o], acc);
  base[b * OUT_DIM + o] = acc;
}

// ---------------------------------------------------------------------------
// Kernels 3/5: fp32 WMMA GEMM  C[M,N] = A[M,K] * B[K,N]  (row-major, K,N=1024)
// 256 threads = 8 waves, 128x128 block tile, wave tile 32x64 (2x4 frags).
// Both A and B^T staged in LDS with ld=36: multiples-of-4 bank rotation makes
// the b64 fragment reads conflict-free, and 36 keeps b128 store alignment.
// B is fetched column-wise from global (still coalesced per k-row) so the LDS
// tile is transposed and a B fragment is one contiguous ds_load_b64 — no
// packing movs.  K-slabs are register-pipelined; prefetch runs 2 slabs ahead.
// ---------------------------------------------------------------------------
template <bool MASKED>
__global__ __launch_bounds__(256) void gemm_wmma_f32(
    const float* __restrict__ A, const float* __restrict__ Bm,
    float* __restrict__ C, const int K, const int N,
    const int* __restrict__ lengths) {
  constexpr int KB  = 32;
  constexpr int LDS_LD = KB + 4;              // 36: aligned + conflict-free
  __shared__ float As[128 * LDS_LD];          // [m][k]
  __shared__ float Bt[128 * LDS_LD];          // [n][k]  (transposed tile)

  const int tid  = threadIdx.x;
  const int lane = tid & 31;
  const int wave = tid >> 5;
  const int wr   = wave & 3;                  // 4 waves along M (32 rows each)
  const int wc   = wave >> 2;                 // 2 waves along N (64 cols each)
  const int lo   = lane & 15;
  const int hi   = lane >> 4;
  const int tileM = blockIdx.y * 128;
  const int tileN = blockIdx.x * 128;

  // Staging-register geometry (4 segments per thread for each tile)
  const int aRow = tid >> 3;                  // A: 128 rows x 8 float4-cols
  const int aC4  = (tid & 7) << 2;
  const int bN   = tid & 127;                 // B: 128 cols x 8 k-groups
  const int bKg  = (tid >> 7) << 2;           // k-group base of segment 0 (0 or 4)

  v8f acc[2][4];
#pragma unroll
  for (int i = 0; i < 2; ++i)
#pragma unroll
    for (int j = 0; j < 4; ++j)
      acc[i][j] = v8f{0.f, 0.f, 0.f, 0.f, 0.f, 0.f, 0.f, 0.f};

  v4f aReg[4];
  float bReg[4][4];

  // ---- preload slab 0 into registers -------------------------------------
#pragma unroll
  for (int i = 0; i < 4; ++i) {
    aReg[i] = *(const v4f*)(A + (size_t)(tileM + aRow + i * 32) * K + aC4);
#pragma unroll
    for (int j = 0; j < 4; ++j)
      bReg[i][j] = Bm[(size_t)(bKg + i * 8 + j) * N + tileN + bN];
  }

  for (int k0 = 0; k0 < K; k0 += KB) {
    // ---- commit staged registers to LDS ----------------------------------
#pragma unroll
    for (int i = 0; i < 4; ++i) {
      *(v4f*)(&As[(aRow + i * 32) * LDS_LD + aC4]) = aReg[i];
      *(v4f*)(&Bt[bN * LDS_LD + bKg + i * 8]) =
          v4f{bReg[i][0], bReg[i][1], bReg[i][2], bReg[i][3]};
    }
    __syncthreads();

    // ---- issue next slab's global loads (latency hidden by WMMAs) --------
    const int kn = k0 + KB;
    if (kn < K) {
#pragma unroll
      for (int i = 0; i < 4; ++i) {
        aReg[i] = *(const v4f*)(A + (size_t)(tileM + aRow + i * 32) * K + kn + aC4);
#pragma unroll
        for (int j = 0; j < 4; ++j)
          bReg[i][j] = Bm[(size_t)(kn + bKg + i * 8 + j) * N + tileN + bN];
      }
    }
    // prefetch two slabs ahead into L2/L0 (global_prefetch_b8)
    if (k0 + 2 * KB < K) {
      __builtin_prefetch(A + (size_t)(tileM + aRow) * K + k0 + 2 * KB + aC4, 0, 1);
      __builtin_prefetch(Bm + (size_t)(k0 + 2 * KB + bKg) * N + tileN + bN, 0, 1);
    }

    // ---- 64 WMMAs on the resident slab -----------------------------------
#pragma unroll
    for (int kk = 0; kk < KB; kk += 4) {
      v2f afrag[2];
#pragma unroll
      for (int mf = 0; mf < 2; ++mf)
        afrag[mf] = *(const v2f*)(&As[(wr * 32 + mf * 16 + lo) * LDS_LD + kk + 2 * hi]);
      v2f bfrag[4];
#pragma unroll
      for (int nf = 0; nf < 4; ++nf)
        bfrag[nf] = *(const v2f*)(&Bt[(wc * 64 + nf * 16 + lo) * LDS_LD + kk + 2 * hi]);
#pragma unroll
      for (int nf = 0; nf < 4; ++nf)
#pragma unroll
        for (int mf = 0; mf < 2; ++mf)
          acc[mf][nf] = __builtin_amdgcn_wmma_f32_16x16x4_f32(
              false, afrag[mf], false, bfrag[nf], (short)0, acc[mf][nf],
              false, false);
    }
    __syncthreads();
  }

  // ---- epilogue: C/D layout VGPR r -> M = r + 8*hi, N = lo ---------------
#pragma unroll
  for (int mf = 0; mf < 2; ++mf) {
#pragma unroll
    for (int nf = 0; nf < 4; ++nf) {
#pragma unroll
      for (int r = 0; r < 8; ++r) {
        const int m = tileM + wr * 32 + mf * 16 + r + 8 * hi;
        const int n = tileN + wc * 64 + nf * 16 + lo;
        float v = acc[mf][nf][r];
        if (MASKED) {
          const int t = m & (T_DIM - 1);
          const int b = m >> 9;               // T = 512
          if (t >= lengths[b]) v = 0.f;
        }
        C[(size_t)m * N + n] = v;
      }
    }
  }
}

// ---------------------------------------------------------------------------
// Kernel 4: h[b,t,:] = gelu_exact(base[b,:] + P[pos[b,t],:]); also emits
// pred_mask (as float 0/1) into the tail of d_out.
// ---------------------------------------------------------------------------
__global__ __launch_bounds__(256) void hbuild_kernel(
    const float* __restrict__ P, const float* __restrict__ base,
    const int* __restrict__ pos, const int* __restrict__ lengths,
    float* __restrict__ h, float* __restrict__ mask_out) {
  const int o  = blockIdx.x * 256 + threadIdx.x;
  const int bt = blockIdx.y;                  // 0 .. B*T-1
  const int b  = bt >> 9;                     // T = 512
  const int t  = bt & (T_DIM - 1);
  const int p  = pos[bt];
  const float x = base[b * OUT_DIM + o] + P[(size_t)p * OUT_DIM + o];
  const float g = 0.5f * x * (1.0f + erff(x * 0.70710678118654752f));
  h[(size_t)bt * OUT_DIM + o] = g;
  if (blockIdx.x == 0 && threadIdx.x == 0)
    mask_out[bt] = (t < lengths[b]) ? 1.0f : 0.0f;
}

// ---------------------------------------------------------------------------
extern "C" void kernel_launch(void* const* d_in, const int* in_sizes, int n_in,
                              void* d_out, int out_size, void* d_ws,
                              size_t ws_size, hipStream_t stream) {
  const float* ctx      = (const float*)d_in[0];          // [16,2048,1024]
  const float* postab   = (const float*)d_in[1];          // [4096,1024]
  const float* W1       = (const float*)d_in[2];          // [2048,1024]
  const float* W2       = (const float*)d_in[3];          // [1024,1024]
  const int*   tpos     = (const int*)d_in[4];            // [16,512]
  const int*   tlen     = (const int*)d_in[5];            // [16]
  const unsigned char* cmask = (const unsigned char*)d_in[6]; // [16,2048] bool

  float* out_pred = (float*)d_out;                        // [16,512,1024]
  float* out_mask = out_pred + (size_t)B_DIM * T_DIM * OUT_DIM; // [16,512]

  // workspace layout (floats): part | pcnt | pooled | base | P | h  (~51 MB)
  float* ws     = (float*)d_ws;
  float* part   = ws;                                     // 8*16*1024
  float* pcnt   = part + (size_t)ZSPLIT * B_DIM * D_DIM;  // 8*16 (round to 128)
  float* pooled = pcnt + 128;                             // 16*1024
  float* base   = pooled + B_DIM * D_DIM;                 // 16*1024
  float* P      = base + B_DIM * OUT_DIM;                 // 4096*1024
  float* h      = P + (size_t)L_DIM * OUT_DIM;            // 8192*1024

  // 1) masked mean pool (HBM-bound, ~6us floor; 512 blocks for occupancy)
  pool_partial<<<dim3(D_DIM / 256, B_DIM, ZSPLIT), 256, 0, stream>>>(
      ctx, cmask, part, pcnt);
  pool_reduce<<<dim3(D_DIM / 256, B_DIM), 256, 0, stream>>>(part, pcnt, pooled);

  // 2) base = pooled @ W1[:1024,:]
  base_kernel<<<dim3(OUT_DIM / 256, B_DIM), 256, 0, stream>>>(pooled, W1, base);

  // 3) P = pos_table @ W1[1024:,:]   (M=4096) — fp32 WMMA
  gemm_wmma_f32<false><<<dim3(OUT_DIM / 128, L_DIM / 128), 256, 0, stream>>>(
      postab, W1 + (size_t)D_DIM * OUT_DIM, P, D_DIM, OUT_DIM, nullptr);

  // 4) h = gelu(base[b] + P[pos[b,t]]) ; emit pred_mask
  hbuild_kernel<<<dim3(OUT_DIM / 256, B_DIM * T_DIM), 256, 0, stream>>>(
      P, base, tpos, tlen, h, out_mask);

  // 5) predictions = mask * (h @ W2)   (M=8192) — fp32 WMMA, mask in epilogue
  gemm_wmma_f32<true><<<dim3(OUT_DIM / 128, (B_DIM * T_DIM) / 128), 256, 0,
                        stream>>>(h, W2, out_pred, D_DIM, OUT_DIM, tlen);
}